// PointCloudAttention_15676630630788
// MI455X (gfx1250) — compile-verified
//
#include <hip/hip_runtime.h>
#include <math.h>

typedef __attribute__((ext_vector_type(2))) float v2f;
typedef __attribute__((ext_vector_type(8))) float v8f;
typedef __attribute__((ext_vector_type(4))) unsigned int v4u;
typedef __attribute__((ext_vector_type(8))) int v8i;
typedef __attribute__((ext_vector_type(4))) int v4i;

#define NB 8
#define DD 64
#define PP 4096
#define NPD (NB * PP * DD)
#define NP  (NB * PP)
#define QSCALE 0.125f          // 1/sqrt(64)
#define PTILES (PP / 16)       // 256 16-key tiles per batch
#define WTILES (NP / 16)       // 2048 16-row wave tiles total
#define BIN_SIZE 10

#define HAVE_TDM (__has_builtin(__builtin_amdgcn_tensor_load_to_lds) && \
                  __has_builtin(__builtin_amdgcn_s_wait_tensorcnt))

// ---------------------------------------------------------------------------
// f32 WMMA helper: D = A(16x4) * B(4x16) + C(16x16)
// A: lane l holds rows M=l%16, K = 2*(l/16)+{0,1}  (2 VGPRs)
// B: lane l holds col  N=l%16, K = 2*(l/16)+{0,1}  (2 VGPRs)
// C/D: reg r, lane l -> row r+8*(l/16), col l%16   (8 VGPRs)
// ---------------------------------------------------------------------------
__device__ __forceinline__ v8f wmma_f32(v2f a, v2f b, v8f c) {
  return __builtin_amdgcn_wmma_f32_16x16x4_f32(false, a, false, b, (short)0, c,
                                               false, false);
}

__device__ __forceinline__ float hmax16(float v) {
  v = fmaxf(v, __shfl_xor(v, 1));
  v = fmaxf(v, __shfl_xor(v, 2));
  v = fmaxf(v, __shfl_xor(v, 4));
  v = fmaxf(v, __shfl_xor(v, 8));
  return v;
}
__device__ __forceinline__ float hsum16(float v) {
  v += __shfl_xor(v, 1);
  v += __shfl_xor(v, 2);
  v += __shfl_xor(v, 4);
  v += __shfl_xor(v, 8);
  return v;
}

#if HAVE_TDM
// ---------------------------------------------------------------------------
// TDM: 2D tile load Global->LDS per CDNA5 D# descriptor layout (ISA 08 §8).
// group0: [1:0]count=1 | [63:32]lds_addr | [120:64]global_addr | [127:126]type=2
// group1: data_size=2(4B), tensor dims (large), tile_d0/d1, dim0 stride.
// 6-arg builtin (clang-23 form): (g0, g1, g2, g3, g4, cpol)
// ---------------------------------------------------------------------------
__device__ __forceinline__ void tdm_load_2d(unsigned lds_byte_addr, const void* gp,
                                            unsigned tile_d0, unsigned tile_d1,
                                            unsigned stride0) {
  unsigned long long ga = (unsigned long long)(size_t)gp;
  unsigned ga_lo = (unsigned)__builtin_amdgcn_readfirstlane((int)(unsigned)ga);
  unsigned ga_hi = (unsigned)__builtin_amdgcn_readfirstlane((int)(unsigned)(ga >> 32));
  unsigned lds   = (unsigned)__builtin_amdgcn_readfirstlane((int)lds_byte_addr);
  v4u g0;
  g0[0] = 1u;                                   // count=1, user descriptor
  g0[1] = lds;                                  // LDS byte address
  g0[2] = ga_lo;                                // global_addr[31:0]
  g0[3] = (ga_hi & 0x01FFFFFFu) | (2u << 30);   // global_addr[56:32] | type=2
  const unsigned td0 = 1u << 22, td1 = 1u << 22;  // generous bounds: no OOB zeroing
  v8i g1;
  g1[0] = (int)(2u << 16);                                            // data_size=4B
  g1[1] = (int)((td0 & 0xFFFFu) << 16);                               // tensor_dim0 lo
  g1[2] = (int)(((td0 >> 16) & 0xFFFFu) | ((td1 & 0xFFFFu) << 16));   // d0 hi | d1 lo
  g1[3] = (int)(((td1 >> 16) & 0xFFFFu) | (tile_d0 << 16));           // d1 hi | tile_d0
  g1[4] = (int)tile_d1;                                               // tile_d1 | tile_d2=0
  g1[5] = (int)stride0;                                               // dim0 stride lo
  g1[6] = 0;                                                          // stride hi / dim1 stride lo
  g1[7] = 0;
  v4i z4 = {0, 0, 0, 0};
  v8i z8 = {0, 0, 0, 0, 0, 0, 0, 0};
  __builtin_amdgcn_tensor_load_to_lds(g0, g1, z4, z4, z8, 0);
}
#endif

// ---------------------------------------------------------------------------
// K0: init scratch scalars + output counts
// ---------------------------------------------------------------------------
__global__ void k0_init(float* counts, int* gmm) {
  int t = threadIdx.x;
  if (t < BIN_SIZE) counts[t] = 0.0f;
  if (t == 0) { gmm[0] = 0x7f800000; gmm[1] = 0; }  // +inf bits / 0 bits
}

// ---------------------------------------------------------------------------
// K1: q = 0.125 * X^T Wq^T, k = X^T Wk^T (row-major n,p,d), v^T (n,d,p)
// ---------------------------------------------------------------------------
__global__ __launch_bounds__(256) void k1_qkv(
    const float* __restrict__ x, const float* __restrict__ Wv,
    const float* __restrict__ Wk, const float* __restrict__ Wq,
    float* __restrict__ qs, float* __restrict__ ks_, float* __restrict__ vt) {
  int wid  = blockIdx.x * 8 + (threadIdx.x >> 5);
  int lane = threadIdx.x & 31, half = lane >> 4, j = lane & 15;
  int n = wid / PTILES, p0 = (wid % PTILES) * 16;
  const float* xb = x + (size_t)n * DD * PP;
  int pm = p0 + j;

  v2f a[16];
#pragma unroll
  for (int cc = 0; cc < 16; ++cc) {
    int d0 = cc * 4 + half * 2;
    v2f t; t[0] = xb[(size_t)d0 * PP + pm]; t[1] = xb[(size_t)(d0 + 1) * PP + pm];
    a[cc] = t;
  }
#pragma unroll
  for (int w = 0; w < 3; ++w) {
    const float* W = (w == 0) ? Wq : (w == 1) ? Wk : Wv;
#pragma unroll
    for (int t4 = 0; t4 < 4; ++t4) {
      int col = t4 * 16 + j;
      v8f acc = {};
#pragma unroll
      for (int cc = 0; cc < 16; ++cc) {
        v2f b = *reinterpret_cast<const v2f*>(W + (size_t)col * DD + cc * 4 + half * 2);
        acc = wmma_f32(a[cc], b, acc);
      }
#pragma unroll
      for (int r = 0; r < 8; ++r) {
        int p = p0 + r + 8 * half;
        if (w == 0)      qs [(size_t)n * PP * DD + (size_t)p * DD + col] = acc[r] * QSCALE;
        else if (w == 1) ks_[(size_t)n * PP * DD + (size_t)p * DD + col] = acc[r];
        else             vt [(size_t)n * DD * PP + (size_t)col * PP + p] = acc[r];
      }
    }
  }
}

// ---------------------------------------------------------------------------
// K2a: flash attention. Block = 8 waves x 32 queries = 256 queries.
// K/V 16-key tiles staged once per block into LDS via TDM (double-buffered,
// TENSORcnt + workgroup barriers); waves run S-WMMAs, online softmax,
// LDS transpose of P, PV-WMMAs, all from LDS operands.
// ---------------------------------------------------------------------------
__global__ __launch_bounds__(256) void k2a_flash(
    const float* __restrict__ qs, const float* __restrict__ ks_,
    const float* __restrict__ vt, float* __restrict__ ao,
    float* __restrict__ m_arr, float* __restrict__ il_arr) {
  __shared__ float kbuf[2][16 * 64];  // [key][d], tile contiguous in global
  __shared__ float vbuf[2][64 * 16];  // [d][key], rows stride P in global
  __shared__ float plds[8][16 * 18];  // per-wave P-transpose tile (padded)

  int widx = threadIdx.x >> 5;
  int lane = threadIdx.x & 31, half = lane >> 4, j = lane & 15;
  int n  = blockIdx.x / (PP / 256);
  int p0 = (blockIdx.x % (PP / 256)) * 256 + widx * 32;

  const float* kb = ks_ + (size_t)n * PP * DD;
  const float* vb = vt  + (size_t)n * DD * PP;

  // 32 query rows per wave: two 16-row A tiles, kept in registers
  v2f a0[16], a1[16];
  {
    const float* q0 = qs + (size_t)n * PP * DD + (size_t)(p0 + j) * DD;
    const float* q1 = q0 + 16 * DD;
#pragma unroll
    for (int cc = 0; cc < 16; ++cc) {
      a0[cc] = *reinterpret_cast<const v2f*>(q0 + cc * 4 + half * 2);
      a1[cc] = *reinterpret_cast<const v2f*>(q1 + cc * 4 + half * 2);
    }
  }

  v8f out0[4] = {}, out1[4] = {};
  float m0[8], l0[8], m1[8], l1[8];
#pragma unroll
  for (int r = 0; r < 8; ++r) {
    m0[r] = -INFINITY; l0[r] = 0.0f;
    m1[r] = -INFINITY; l1[r] = 0.0f;
  }

#if HAVE_TDM
  // Prime double buffer: tiles 0 and 1 (2 TDM ops each).
  if (widx == 0) {
    tdm_load_2d((unsigned)(size_t)&kbuf[0][0], kb,            64, 16, 64);
    tdm_load_2d((unsigned)(size_t)&vbuf[0][0], vb,            16, 64, PP);
    tdm_load_2d((unsigned)(size_t)&kbuf[1][0], kb + 16 * DD,  64, 16, 64);
    tdm_load_2d((unsigned)(size_t)&vbuf[1][0], vb + 16,       16, 64, PP);
    __builtin_amdgcn_s_wait_tensorcnt(2);  // tile0 pair complete
  }
  __syncthreads();
#endif

  for (int kt = 0; kt < PTILES; ++kt) {
    int kbase = kt * 16;
    int cur = kt & 1;
#if !HAVE_TDM
    // Fallback: cooperative staging of K/V tiles (2048 floats, 8 per thread)
    __syncthreads();
    {
      const float* gk = kb + (size_t)kbase * DD;
      const float* gv = vb + kbase;
      for (int i = threadIdx.x; i < 1024; i += 256) kbuf[cur][i] = gk[i];
      for (int i = threadIdx.x; i < 1024; i += 256)
        vbuf[cur][i] = gv[(size_t)(i >> 4) * PP + (i & 15)];
    }
    __syncthreads();
#endif
    // ---- S = q * k^T for both 16-row query tiles (K tile shared from LDS)
    v8f s0 = {}, s1 = {};
#pragma unroll
    for (int cc = 0; cc < 16; ++cc) {
      v2f bk = *reinterpret_cast<const v2f*>(&kbuf[cur][j * 64 + cc * 4 + half * 2]);
      s0 = wmma_f32(a0[cc], bk, s0);
      s1 = wmma_f32(a1[cc], bk, s1);
    }
    // ---- online softmax, both tiles
    float p0v[8], p1v[8];
#pragma unroll
    for (int r = 0; r < 8; ++r) {
      float nm = fmaxf(m0[r], hmax16(s0[r]));
      float sc = __expf(m0[r] - nm);
      float pv = __expf(s0[r] - nm);
      l0[r] = l0[r] * sc + hsum16(pv);
      m0[r] = nm; p0v[r] = pv;
      out0[0][r] *= sc; out0[1][r] *= sc; out0[2][r] *= sc; out0[3][r] *= sc;
    }
#pragma unroll
    for (int r = 0; r < 8; ++r) {
      float nm = fmaxf(m1[r], hmax16(s1[r]));
      float sc = __expf(m1[r] - nm);
      float pv = __expf(s1[r] - nm);
      l1[r] = l1[r] * sc + hsum16(pv);
      m1[r] = nm; p1v[r] = pv;
      out1[0][r] *= sc; out1[1][r] *= sc; out1[2][r] *= sc; out1[3][r] *= sc;
    }
    // ---- transpose P tiles (C layout -> A layout) via per-wave LDS region.
    // Same-wave LDS ops are in-order; region reused sequentially for both tiles.
    v2f pa0[4], pa1[4];
#pragma unroll
    for (int r = 0; r < 8; ++r) plds[widx][(r + 8 * half) * 18 + j] = p0v[r];
    asm volatile("s_wait_dscnt 0" ::: "memory");
#pragma unroll
    for (int c2 = 0; c2 < 4; ++c2)
      pa0[c2] = *reinterpret_cast<const v2f*>(&plds[widx][j * 18 + c2 * 4 + half * 2]);
#pragma unroll
    for (int r = 0; r < 8; ++r) plds[widx][(r + 8 * half) * 18 + j] = p1v[r];
    asm volatile("s_wait_dscnt 0" ::: "memory");
#pragma unroll
    for (int c2 = 0; c2 < 4; ++c2)
      pa1[c2] = *reinterpret_cast<const v2f*>(&plds[widx][j * 18 + c2 * 4 + half * 2]);
    // ---- out += P * V (V tile shared from LDS; each bv feeds both tiles)
#pragma unroll
    for (int t4 = 0; t4 < 4; ++t4) {
#pragma unroll
      for (int c2 = 0; c2 < 4; ++c2) {
        v2f bv = *reinterpret_cast<const v2f*>(
            &vbuf[cur][(t4 * 16 + j) * 16 + c2 * 4 + half * 2]);
        out0[t4] = wmma_f32(pa0[c2], bv, out0[t4]);
        out1[t4] = wmma_f32(pa1[c2], bv, out1[t4]);
      }
    }
#if HAVE_TDM
    __syncthreads();  // all waves done reading buf[cur]
    if (widx == 0) {
      if (kt + 2 < PTILES) {
        int nb = (kt + 2) * 16;
        tdm_load_2d((unsigned)(size_t)&kbuf[cur][0], kb + (size_t)nb * DD, 64, 16, 64);
        tdm_load_2d((unsigned)(size_t)&vbuf[cur][0], vb + nb,              16, 64, PP);
        __builtin_amdgcn_s_wait_tensorcnt(2);  // pair (kt+1) complete
      } else {
        __builtin_amdgcn_s_wait_tensorcnt(0);
      }
    }
    __syncthreads();  // buf[(kt+1)&1] ready for all waves
#endif
  }
  // ---- epilogue: normalize, store attn-out, m, 1/l for both tiles
  float il0[8], il1[8];
#pragma unroll
  for (int r = 0; r < 8; ++r) { il0[r] = 1.0f / l0[r]; il1[r] = 1.0f / l1[r]; }
#pragma unroll
  for (int t4 = 0; t4 < 4; ++t4)
#pragma unroll
    for (int r = 0; r < 8; ++r) {
      size_t base = (size_t)n * PP * DD;
      ao[base + (size_t)(p0 + r + 8 * half) * DD + t4 * 16 + j]      = out0[t4][r] * il0[r];
      ao[base + (size_t)(p0 + 16 + r + 8 * half) * DD + t4 * 16 + j] = out1[t4][r] * il1[r];
    }
  if (j == 0) {
#pragma unroll
    for (int r = 0; r < 8; ++r) {
      int i0 = n * PP + p0 + r + 8 * half;
      m_arr[i0]       = m0[r];  il_arr[i0]      = il0[r];
      m_arr[i0 + 16]  = m1[r];  il_arr[i0 + 16] = il1[r];
    }
  }
}

// ---------------------------------------------------------------------------
// K2b: max over queries per key: recompute S^T (A = K rows, B = Q cols),
// p = exp(s - m_q) / l_q, register max-reduce -> maxatt[n][key]. No atomics.
// ---------------------------------------------------------------------------
__global__ __launch_bounds__(256) void k2b_colmax(
    const float* __restrict__ qs, const float* __restrict__ ks_,
    const float* __restrict__ m_arr, const float* __restrict__ il_arr,
    float* __restrict__ maxatt) {
  int wid  = blockIdx.x * 8 + (threadIdx.x >> 5);
  int lane = threadIdx.x & 31, half = lane >> 4, j = lane & 15;
  int n = wid / PTILES, k0 = (wid % PTILES) * 16;

  const float* krow = ks_ + (size_t)n * PP * DD + (size_t)(k0 + j) * DD;
  v2f a[16];
#pragma unroll
  for (int cc = 0; cc < 16; ++cc)
    a[cc] = *reinterpret_cast<const v2f*>(krow + cc * 4 + half * 2);

  float acc[8];
#pragma unroll
  for (int r = 0; r < 8; ++r) acc[r] = 0.0f;

  for (int qt = 0; qt < PTILES; ++qt) {
    int qb = qt * 16;
    int qi = n * PP + qb + j;            // this lane's query column
    float mj  = m_arr[qi];
    float ilj = il_arr[qi];
    const float* qrow = qs + (size_t)n * PP * DD + (size_t)(qb + j) * DD;
    v8f s = {};
#pragma unroll
    for (int cc = 0; cc < 16; ++cc) {
      v2f b = *reinterpret_cast<const v2f*>(qrow + cc * 4 + half * 2);
      s = wmma_f32(a[cc], b, s);
    }
#pragma unroll
    for (int r = 0; r < 8; ++r)
      acc[r] = fmaxf(acc[r], __expf(s[r] - mj) * ilj);
  }
#pragma unroll
  for (int r = 0; r < 8; ++r) {
    float v = hmax16(acc[r]);
    if (j == 0) maxatt[n * PP + k0 + r + 8 * half] = v;
  }
}

// ---------------------------------------------------------------------------
// K3: out = attn_out @ Wout^T + bout
// ---------------------------------------------------------------------------
__global__ __launch_bounds__(256) void k3_outproj(
    const float* __restrict__ ao, const float* __restrict__ Wout,
    const float* __restrict__ bout, float* __restrict__ outp) {
  int wid  = blockIdx.x * 8 + (threadIdx.x >> 5);
  int lane = threadIdx.x & 31, half = lane >> 4, j = lane & 15;
  int n = wid / PTILES, p0 = (wid % PTILES) * 16;

  const float* arow = ao + (size_t)n * PP * DD + (size_t)(p0 + j) * DD;
  v2f a[16];
#pragma unroll
  for (int cc = 0; cc < 16; ++cc)
    a[cc] = *reinterpret_cast<const v2f*>(arow + cc * 4 + half * 2);
#pragma unroll
  for (int t4 = 0; t4 < 4; ++t4) {
    int col = t4 * 16 + j;
    v8f acc = {};
#pragma unroll
    for (int cc = 0; cc < 16; ++cc) {
      v2f b = *reinterpret_cast<const v2f*>(Wout + (size_t)col * DD + cc * 4 + half * 2);
      acc = wmma_f32(a[cc], b, acc);
    }
    float bias = bout[col];
#pragma unroll
    for (int r = 0; r < 8; ++r)
      outp[(size_t)n * PP * DD + (size_t)(p0 + r + 8 * half) * DD + col] =
          acc[r] + bias;
  }
}

// ---------------------------------------------------------------------------
// K4: per-batch min/max of maxatt -> voxel sizes; global voxel min/max (int-bit
// atomics are order-correct: all values positive)
// ---------------------------------------------------------------------------
__global__ __launch_bounds__(256) void k4_voxel(
    const float* __restrict__ maxatt, float* __restrict__ vox, int* gmm) {
  __shared__ float smn[256], smx[256];
  int n = blockIdx.x, t = threadIdx.x;
  float mn = 1e30f, mx = -1e30f;
  for (int i = t; i < PP; i += 256) {
    float v = maxatt[n * PP + i];
    mn = fminf(mn, v); mx = fmaxf(mx, v);
  }
  smn[t] = mn; smx[t] = mx; __syncthreads();
  for (int s = 128; s > 0; s >>= 1) {
    if (t < s) { smn[t] = fminf(smn[t], smn[t + s]); smx[t] = fmaxf(smx[t], smx[t + s]); }
    __syncthreads();
  }
  mn = smn[0]; mx = smx[0];
  float inv = 1.0f / (mx - mn);
  float vmn = 1e30f, vmx = -1e30f;
  __syncthreads();
  for (int i = t; i < PP; i += 256) {
    float norm = (maxatt[n * PP + i] - mn) * inv;
    float v = 0.05f + (1.0f - norm) * 0.1f;
    vox[n * PP + i] = v;
    vmn = fminf(vmn, v); vmx = fmaxf(vmx, v);
  }
  smn[t] = vmn; smx[t] = vmx; __syncthreads();
  for (int s = 128; s > 0; s >>= 1) {
    if (t < s) { smn[t] = fminf(smn[t], smn[t + s]); smx[t] = fmaxf(smx[t], smx[t + s]); }
    __syncthreads();
  }
  if (t == 0) {
    atomicMin(gmm + 0, __float_as_int(smn[0]));
    atomicMax(gmm + 1, __float_as_int(smx[0]));
  }
}

// ---------------------------------------------------------------------------
// K5: bucketize + histogram counts
// ---------------------------------------------------------------------------
__global__ __launch_bounds__(256) void k5_bins(
    const float* __restrict__ vox, const int* __restrict__ gmm,
    int* __restrict__ bidx, float* __restrict__ counts) {
  int i = blockIdx.x * blockDim.x + threadIdx.x;
  if (i >= NP) return;
  float gmin = __int_as_float(gmm[0]);
  float gmax = __int_as_float(gmm[1]);
  float step = (gmax - gmin) / (float)BIN_SIZE;
  int bi = (int)floorf((vox[i] - gmin) / step);
  if (bi < 0 || bi >= BIN_SIZE) bi = BIN_SIZE - 1;
  bidx[i] = bi;
  atomicAdd(counts + bi, 1.0f);
}

// ---------------------------------------------------------------------------
extern "C" void kernel_launch(void* const* d_in, const int* in_sizes, int n_in,
                              void* d_out, int out_size, void* d_ws, size_t ws_size,
                              hipStream_t stream) {
  const float* x    = (const float*)d_in[0];  // (N, D, P)
  const float* Wv   = (const float*)d_in[1];
  const float* Wk   = (const float*)d_in[2];
  const float* Wq   = (const float*)d_in[3];
  const float* Wout = (const float*)d_in[4];
  const float* bout = (const float*)d_in[5];

  // workspace layout (floats)
  float* w      = (float*)d_ws;
  float* qs     = w;                    // N*P*D (pre-scaled q)
  float* ks_    = w + (size_t)NPD;      // N*P*D
  float* vt     = w + 2 * (size_t)NPD;  // N*D*P (v transposed)
  float* ao     = w + 3 * (size_t)NPD;  // N*P*D attention output (pre Wout)
  float* m_arr  = w + 4 * (size_t)NPD;  // N*P
  float* il_arr = m_arr + NP;           // N*P
  float* maxatt = il_arr + NP;          // N*P
  int*   gmm    = (int*)(maxatt + NP);  // [gmin_bits, gmax_bits]

  // output layout: out | voxel_sizes | bin_idx(int32) | counts
  float* outp   = (float*)d_out;
  float* vox    = outp + (size_t)NPD;
  int*   bidx   = (int*)(vox + NP);
  float* counts = (float*)(bidx + NP);

  const int blocks16 = WTILES / 8;  // 8 waves (wave32) per 256-thread block

  k0_init<<<1, 64, 0, stream>>>(counts, gmm);
  k1_qkv<<<blocks16, 256, 0, stream>>>(x, Wv, Wk, Wq, qs, ks_, vt);
  k2a_flash<<<NP / 256, 256, 0, stream>>>(qs, ks_, vt, ao, m_arr, il_arr);
  k2b_colmax<<<blocks16, 256, 0, stream>>>(qs, ks_, m_arr, il_arr, maxatt);
  k3_outproj<<<blocks16, 256, 0, stream>>>(ao, Wout, bout, outp);
  k4_voxel<<<NB, 256, 0, stream>>>(maxatt, vox, gmm);
  k5_bins<<<(NP + 255) / 256, 256, 0, stream>>>(vox, gmm, bidx, counts);
}